// SA_72344429133991
// MI455X (gfx1250) — compile-verified
//
#include <hip/hip_runtime.h>

typedef float v2f __attribute__((ext_vector_type(2)));
typedef float v8f __attribute__((ext_vector_type(8)));

// ---------------------------------------------------------------------------
// Kernel 1: per (b,c) 64x64 tile -> row means (over W) into A, col means
// (over H) into Bv.  grid = 256 blocks (b*64+c), 256 threads.
// ---------------------------------------------------------------------------
__global__ void k_reduce(const float* __restrict__ x,
                         float* __restrict__ A, float* __restrict__ Bv) {
    __shared__ float tile[64 * 65];            // pad stride 65: conflict-free cols
    const float* xp = x + (size_t)blockIdx.x * 4096;
    int t = threadIdx.x;
    for (int it = 0; it < 16; ++it) {
        int p = it * 256 + t;                  // coalesced
        tile[(p >> 6) * 65 + (p & 63)] = xp[p];
    }
    __syncthreads();
    if (t < 64) {                              // row mean (over w) -> A
        float s = 0.f;
        #pragma unroll 8
        for (int w = 0; w < 64; ++w) s += tile[t * 65 + w];
        A[blockIdx.x * 64 + t] = s * (1.0f / 64.0f);
    } else if (t < 128) {                      // col mean (over h) -> Bv
        int w = t - 64;
        float s = 0.f;
        #pragma unroll 8
        for (int h = 0; h < 64; ++h) s += tile[h * 65 + w];
        Bv[blockIdx.x * 64 + w] = s * (1.0f / 64.0f);
    }
}

// ---------------------------------------------------------------------------
// Kernel 2: per-batch max/min of A (for rank-1 softmax stabilization).
// grid = 4 blocks, 256 threads.
// ---------------------------------------------------------------------------
__global__ void k_stats(const float* __restrict__ A, float* __restrict__ stats) {
    __shared__ float smax[256], smin[256];
    const float* a = A + blockIdx.x * 4096;
    int t = threadIdx.x;
    float mx = -__builtin_inff(), mn = __builtin_inff();
    for (int p = t; p < 4096; p += 256) {
        float v = a[p];
        mx = fmaxf(mx, v);
        mn = fminf(mn, v);
    }
    smax[t] = mx; smin[t] = mn;
    __syncthreads();
    for (int s = 128; s > 0; s >>= 1) {
        if (t < s) {
            smax[t] = fmaxf(smax[t], smax[t + s]);
            smin[t] = fminf(smin[t], smin[t + s]);
        }
        __syncthreads();
    }
    if (t == 0) { stats[blockIdx.x * 2] = smax[0]; stats[blockIdx.x * 2 + 1] = smin[0]; }
}

// ---------------------------------------------------------------------------
// Kernel 3 (fused main): one block per (c2, b); 64 columns j = c2*64 + w.
//   Pass A: S[c1][w] = sum_h' exp(a[c1*64+h']*b_j - m_j)   (TRANS pipe)
//   Pass B: T[w]     = ones(1x64) x S  via V_WMMA_F32_16X16X4_F32 (wave 0)
//   Pass C: out[b,c1,c2,w] = x[b,c1,c2,w] + S[c1][w] / (64*T[w])
// ---------------------------------------------------------------------------
__global__ void k_main(const float* __restrict__ x, const float* __restrict__ A,
                       const float* __restrict__ Bv, const float* __restrict__ stats,
                       float* __restrict__ out) {
    __shared__ __align__(16) float aLds[4096];   // a[b, 0..4095]
    __shared__ float sLds[4096];                 // S[c1][w], stride 64
    __shared__ float tInv[64];                   // 1/(64*T[w])
    int c2 = blockIdx.x;       // 0..63
    int b  = blockIdx.y;       // 0..3
    int t  = threadIdx.x;      // 0..255

    const float* Ab = A + b * 4096;
    for (int it = 0; it < 4; ++it) {             // 16B vector loads, coalesced
        int p = it * 256 + t;
        ((float4*)aLds)[p] = ((const float4*)Ab)[p];
    }
    float maxA = stats[b * 2], minA = stats[b * 2 + 1];
    int   w  = t & 63;                           // fixed per thread
    float bw = Bv[b * 4096 + c2 * 64 + w];
    float mj = (bw >= 0.f) ? bw * maxA : bw * minA;
    __syncthreads();

    // Pass A: each thread owns (c1, w) pairs; lanes of a wave share c1 ->
    // aLds reads are broadcast (conflict-free); sLds stores are consecutive.
    for (int it = 0; it < 16; ++it) {
        int c1 = it * 4 + (t >> 6);
        const float* ap = &aLds[c1 * 64];
        float s = 0.f;
        #pragma unroll 8
        for (int h = 0; h < 64; ++h)
            s += __expf(ap[h] * bw - mj);
        sLds[c1 * 64 + w] = s;
    }
    __syncthreads();

    // Pass B: wave 0 only (EXEC all-ones within the wave).  D = ones(16x4) x
    // S-chunk(4x16) accumulated over 16 chunks => every row of D = column sums.
    if (t < 32) {
        int lane = t, half = lane >> 4, n = lane & 15;
        v2f ones; ones[0] = 1.0f; ones[1] = 1.0f;
        for (int wt = 0; wt < 4; ++wt) {
            v8f acc = {};
            #pragma unroll
            for (int ch = 0; ch < 16; ++ch) {
                v2f bv;
                bv[0] = sLds[(ch * 4 + half)     * 64 + wt * 16 + n];
                bv[1] = sLds[(ch * 4 + 2 + half) * 64 + wt * 16 + n];
                acc = __builtin_amdgcn_wmma_f32_16x16x4_f32(
                    /*neg_a=*/false, ones, /*neg_b=*/false, bv,
                    /*c_mod=*/(short)0, acc, /*reuse_a=*/false, /*reuse_b=*/false);
            }
            float T = acc[0];                    // D[0][n] == T[wt*16+n]
            if (lane < 16) tInv[wt * 16 + n] = 1.0f / (64.0f * T);
        }
    }
    __syncthreads();

    // Pass C: out[b,c1,c2,w] = x + S/(64*T).  w fastest -> coalesced 128B/wave.
    float tv = tInv[w];
    for (int it = 0; it < 16; ++it) {
        int c1 = it * 4 + (t >> 6);
        size_t idx = ((((size_t)b * 64 + c1) * 64) + c2) * 64 + w;
        out[idx] = x[idx] + sLds[c1 * 64 + w] * tv;
    }
}

// ---------------------------------------------------------------------------
extern "C" void kernel_launch(void* const* d_in, const int* in_sizes, int n_in,
                              void* d_out, int out_size, void* d_ws, size_t ws_size,
                              hipStream_t stream) {
    (void)in_sizes; (void)n_in; (void)out_size; (void)ws_size;
    const float* x = (const float*)d_in[0];
    float* out = (float*)d_out;

    float* A     = (float*)d_ws;      // 4*4096 floats
    float* Bv    = A + 4 * 4096;      // 4*4096 floats
    float* stats = Bv + 4 * 4096;     // 4*2 floats

    k_reduce<<<dim3(256), 256, 0, stream>>>(x, A, Bv);
    k_stats <<<dim3(4),   256, 0, stream>>>(A, stats);
    k_main  <<<dim3(64, 4), 256, 0, stream>>>(x, A, Bv, stats, out);
}